// ContiguousNormalizedFlow_28552942584189
// MI455X (gfx1250) — compile-verified
//
#include <hip/hip_runtime.h>
#include <math.h>

#define FWIDTH 64
#define HID 64

typedef float v2f __attribute__((ext_vector_type(2)));
typedef float v4f __attribute__((ext_vector_type(4)));
typedef float v8f __attribute__((ext_vector_type(8)));

__device__ __forceinline__ float dtanh_f32(float x) {
#if __has_builtin(__builtin_amdgcn_tanhf)
    return __builtin_amdgcn_tanhf(x);   // v_tanh_f32 (gfx1250 TRANS op)
#else
    return tanhf(x);
#endif
}

// Workspace layout (floats):
//   [0   .. 255]  A-tiles for WMMA: ws[t*64 + part*32 + lane]
//                 part 0 -> A VGPR0 (K0 for lanes 0-15, K2 for lanes 16-31)
//                 part 1 -> A VGPR1 (K1 for lanes 0-15, K3 for lanes 16-31)
//   [256 .. 383]  interleaved u pairs: ws[256+2k] = u0[k], ws[256+2k+1] = u1[k]
//   [384 .. 447]  wu[k] = w_k . u_k

// ---------------------------------------------------------------------------
// Stage 1: scalar-t hypernetwork (1 workgroup, 64 threads). Tiny.
// ---------------------------------------------------------------------------
__global__ void hypernet_kernel(const float* __restrict__ t,
                                const float* __restrict__ W1, const float* __restrict__ b1,
                                const float* __restrict__ W2, const float* __restrict__ b2,
                                const float* __restrict__ Wu, const float* __restrict__ bu,
                                const float* __restrict__ Ww, const float* __restrict__ bw,
                                const float* __restrict__ Wb, const float* __restrict__ bb,
                                float* __restrict__ ws) {
    __shared__ float mo1[HID];
    __shared__ float mo2[HID];
    __shared__ float u_s[FWIDTH * 2];   // u_flat
    __shared__ float w_s[FWIDTH * 2];   // w_flat
    __shared__ float b_s[FWIDTH];

    const int h = threadIdx.x;          // 0..63
    const float tv = t[0];

    mo1[h] = dtanh_f32(tv * W1[h] + b1[h]);
    __syncthreads();

    {
        float acc = b2[h];
        for (int j = 0; j < HID; ++j) acc += mo1[j] * W2[j * HID + h];
        mo2[h] = dtanh_f32(acc);
    }
    __syncthreads();

    {   // heads: each thread produces 2 u, 2 w, 1 b outputs
        float a0 = bu[h], a1 = bu[h + FWIDTH];
        float c0 = bw[h], c1 = bw[h + FWIDTH];
        float e0 = bb[h];
        for (int j = 0; j < HID; ++j) {
            float m = mo2[j];
            a0 += m * Wu[j * (FWIDTH * 2) + h];
            a1 += m * Wu[j * (FWIDTH * 2) + h + FWIDTH];
            c0 += m * Ww[j * (FWIDTH * 2) + h];
            c1 += m * Ww[j * (FWIDTH * 2) + h + FWIDTH];
            e0 += m * Wb[j * FWIDTH + h];
        }
        u_s[h] = a0; u_s[h + FWIDTH] = a1;
        w_s[h] = c0; w_s[h + FWIDTH] = c1;
        b_s[h] = e0;
    }
    __syncthreads();

    {   // per-unit weight tables (h = unit index k): interleaved u pairs + w.u
        float u0 = u_s[2 * h], u1 = u_s[2 * h + 1];
        float w0 = w_s[2 * h], w1 = w_s[2 * h + 1];
        ws[256 + 2 * h]     = u0;
        ws[256 + 2 * h + 1] = u1;
        ws[384 + h]         = w0 * u0 + w1 * u1;
    }

    {   // pre-swizzled WMMA A-tiles: A[m,K] = [w0[m], w1[m], b[m], 0]
        const int part = h >> 5;        // 0 -> A VGPR0, 1 -> A VGPR1
        const int lane = h & 31;
        const int row  = lane & 15;     // M within tile
        for (int tt = 0; tt < 4; ++tt) {
            const int m = tt * 16 + row;
            float v;
            if (lane < 16) v = (part == 0) ? w_s[2 * m] : w_s[2 * m + 1];  // K0 / K1
            else           v = (part == 0) ? b_s[m]     : 0.0f;            // K2 / K3
            ws[tt * 64 + part * 32 + lane] = v;
        }
    }
}

// ---------------------------------------------------------------------------
// Stage 2: per-point flow. One wave handles 16 points per chunk via
// 4x V_WMMA_F32_16X16X4_F32 (units x points), v_tanh + pk_fma reductions,
// with the z-load software-pipelined one chunk ahead.
// ---------------------------------------------------------------------------
__global__ void __launch_bounds__(256) flow_kernel(const float* __restrict__ z,
                                                   const float* __restrict__ ws,
                                                   float* __restrict__ out,
                                                   int Bn, int nChunks) {
    const int lane = threadIdx.x & 31;
    const int wave = (blockIdx.x * blockDim.x + threadIdx.x) >> 5;
    const int nW   = (gridDim.x * blockDim.x) >> 5;
    const int half = lane >> 4;     // which unit half (M 0-7 vs 8-15) this lane holds
    const int col  = lane & 15;     // point column within tile
    const int zMax = 2 * Bn - 1;    // clamp bound for branch-free cooperative loads

    if (Bn <= 0) return;

    // Preload wave-uniform A tiles (w0,w1,b,0 per 16 units).
    v2f A[4];
#pragma unroll
    for (int t = 0; t < 4; ++t) {
        A[t].x = ws[t * 64 + lane];
        A[t].y = ws[t * 64 + 32 + lane];
    }

    // Preload per-lane unit weights matching the WMMA C/D layout:
    // D VGPR r of tile t holds unit k = t*16 + half*8 + r for this lane.
    // Scale factors (1/WIDTH, -1/WIDTH) are folded in here.
    const float inv = 1.0f / (float)FWIDTH;
    v2f   u01v[4][8];               // {u0[k], u1[k]} * inv  -> v_pk_fma_f32 operand
    float wuv[4][8];                // wu[k] * -inv
#pragma unroll
    for (int t = 0; t < 4; ++t) {
        const int k  = t * 16 + half * 8;
        const int kb = 256 + 2 * k;                         // 64B-aligned
        const v4f p0 = *(const v4f*)(ws + kb);
        const v4f p1 = *(const v4f*)(ws + kb + 4);
        const v4f p2 = *(const v4f*)(ws + kb + 8);
        const v4f p3 = *(const v4f*)(ws + kb + 12);
        const v4f wa = *(const v4f*)(ws + 384 + k);         // 32B-aligned
        const v4f wb = *(const v4f*)(ws + 384 + k + 4);
#pragma unroll
        for (int r = 0; r < 2; ++r) {
            u01v[t][r]     = (v2f){p0[2 * r] * inv, p0[2 * r + 1] * inv};
            u01v[t][r + 2] = (v2f){p1[2 * r] * inv, p1[2 * r + 1] * inv};
            u01v[t][r + 4] = (v2f){p2[2 * r] * inv, p2[2 * r + 1] * inv};
            u01v[t][r + 6] = (v2f){p3[2 * r] * inv, p3[2 * r + 1] * inv};
        }
#pragma unroll
        for (int r = 0; r < 4; ++r) {
            wuv[t][r]     = wa[r] * -inv;
            wuv[t][r + 4] = wb[r] * -inv;
        }
    }

    if (wave >= nChunks) return;

    // Cooperative, branch-free z fetch: lane L holds z[chunk*32 + L]
    // (the 16-point tile's 32 floats). Running index, clamped in-bounds.
    const int giStride = nW * 32;
    int gi = wave * 32 + lane;
    if (gi > zMax) gi = zMax;
    float zv = z[gi];                           // prologue load

    for (int chunk = wave; chunk < nChunks; chunk += nW) {
        // Issue next chunk's load immediately (hidden behind this chunk's math).
        gi += giStride;
        if (gi > zMax) gi = zMax;               // v_min: safe, result unused past tail
        const float zvNext = z[gi];

        // Build B matrix (K4 x 16 points): rows [z0, z1, 1, 0].
        // z0[col] lives in lane 2*col, z1[col] in lane 2*col+1.
        const float z0 = __shfl(zv, 2 * col, 32);       // ds_bpermute
        const float z1 = __shfl(zv, 2 * col + 1, 32);
        v2f Bv;
        Bv.x = half ? 1.0f : z0;                        // K2 row = 1 (bias)
        Bv.y = half ? 0.0f : z1;                        // K3 row = 0

        // lin tiles: 4 x (16 units x 16 points)
        v8f d[4];
#pragma unroll
        for (int t = 0; t < 4; ++t) {
            v8f c = {};
            d[t] = __builtin_amdgcn_wmma_f32_16x16x4_f32(
                false, A[t], false, Bv, (short)0, c, false, false);
        }

        // tanh + reductions: acc01 via v_pk_fma_f32 (th splat by op_sel),
        // trace term via scalar fma chain (dual-issues against pk ops).
        v2f acc01 = {0.0f, 0.0f};
        float acc2 = 0.0f;
#pragma unroll
        for (int t = 0; t < 4; ++t)
#pragma unroll
            for (int r = 0; r < 8; ++r) {
                const float th = dtanh_f32(d[t][r]);
                const v2f thv = {th, th};
                acc01 += u01v[t][r] * thv;              // v_pk_fma_f32
                acc2  += (1.0f - th * th) * wuv[t][r];  // v_fma + v_fmac
            }

        // combine the two lane-halves (units 0-7 vs 8-15 of each tile)
        acc01.x += __shfl_xor(acc01.x, 16, 32);
        acc01.y += __shfl_xor(acc01.y, 16, 32);
        acc2    += __shfl_xor(acc2,    16, 32);

        // balanced stores: lanes 0-15 write the dz_dt pair (b64),
        // lanes 16-31 write dlog_p_dt (b32).
        const int p = chunk * 16 + col;
        if (p < Bn) {
            if (half == 0) {
                *(v2f*)(out + 2 * p) = acc01;           // dz_dt[p, 0:2]
            } else {
                out[2 * Bn + p] = acc2;                 // dlog_p_dt[p] (sign folded)
            }
        }

        zv = zvNext;
    }
}

extern "C" void kernel_launch(void* const* d_in, const int* in_sizes, int n_in,
                              void* d_out, int out_size, void* d_ws, size_t ws_size,
                              hipStream_t stream) {
    const float* t  = (const float*)d_in[0];
    const float* z  = (const float*)d_in[1];
    // d_in[2] (dlogp) is unused by the reference outputs
    const float* W1 = (const float*)d_in[3];
    const float* b1 = (const float*)d_in[4];
    const float* W2 = (const float*)d_in[5];
    const float* b2 = (const float*)d_in[6];
    const float* Wu = (const float*)d_in[7];
    const float* bu = (const float*)d_in[8];
    const float* Ww = (const float*)d_in[9];
    const float* bw = (const float*)d_in[10];
    const float* Wb = (const float*)d_in[11];
    const float* bb = (const float*)d_in[12];

    float* ws  = (float*)d_ws;
    float* out = (float*)d_out;
    const int Bn = in_sizes[2];              // dlogp has B elements

    hipLaunchKernelGGL(hypernet_kernel, dim3(1), dim3(64), 0, stream,
                       t, W1, b1, W2, b2, Wu, bu, Ww, bw, Wb, bb, ws);

    const int nChunks = (Bn + 15) / 16;
    int blocks = 512;                        // 4096 waves -> ~15 chunks each
    const int maxBlocks = (nChunks + 7) / 8; // 8 waves per block
    if (blocks > maxBlocks) blocks = maxBlocks;
    if (blocks < 1) blocks = 1;

    hipLaunchKernelGGL(flow_kernel, dim3(blocks), dim3(256), 0, stream,
                       z, ws, out, Bn, nChunks);
}